// DynamicLinearModel_40733469835602
// MI455X (gfx1250) — compile-verified
//
#include <hip/hip_runtime.h>
#include <math.h>

typedef __attribute__((ext_vector_type(2))) float v2f;
typedef __attribute__((ext_vector_type(8))) float v8f;

#define T_TOTAL 1048576
#define DIM     64
#define CHUNK   8192
#define NCH     128      // T_TOTAL / CHUNK
#define LSEG    32       // CHUNK / 256 threads

__device__ __forceinline__ float sigmoidf_(float x) { return 1.f / (1.f + __expf(-x)); }

// ---------------------------------------------------------------------------
// K1: streaming matvec via V_WMMA_F32_16X16X4_F32.
// Per 16-row tile: D = X_tile(16x64) @ Wx + Z_tile(16x64) @ Wz  (32 WMMAs, K=4)
//   Wx col0 = eta  -> D[:,0] = xdot
//   Wz col1 = zeta/2 -> D[:,1] = c
// A f32 16x4 layout: lane(g = lane>>4, r = lane&15) holds A[r][K0+2g], A[r][K0+2g+1]
// B f32 4x16 layout: lane(g,r) holds B[2g][r], B[2g+1][r]
// D f32 16x16 layout: lane(g,r) acc[v] = D[v + 8g][r]
// ---------------------------------------------------------------------------
__global__ __launch_bounds__(256) void k1_matvec_wmma(
    const float* __restrict__ X, const float* __restrict__ Z,
    const float* __restrict__ eta, const float* __restrict__ zeta,
    float* __restrict__ xdot_out /* = d_out[0..T) */,
    float* __restrict__ c_out    /* = ws[0..T)   */)
{
    const int lane = threadIdx.x & 31;
    const int wave = threadIdx.x >> 5;
    const int g = lane >> 4;
    const int r = lane & 15;

    // Build weight (B-matrix) registers once per wave; reused for 16 tiles.
    v2f wx[16], wz[16];
#pragma unroll
    for (int s = 0; s < 16; ++s) {
        v2f e = {0.f, 0.f};
        v2f z = {0.f, 0.f};
        if (r == 0) e = *(const v2f*)(eta + 4 * s + 2 * g);
        if (r == 1) {
            v2f zz = *(const v2f*)(zeta + 4 * s + 2 * g);
            z.x = 0.5f * zz.x; z.y = 0.5f * zz.y;
        }
        wx[s] = e; wz[s] = z;
    }

    const int waveGlobal = blockIdx.x * 8 + wave;   // 512 blocks * 8 waves = 4096
    for (int it = 0; it < 16; ++it) {
        const int tile = waveGlobal * 16 + it;       // 65536 tiles of 16 rows
        const int row0 = tile * 16;
        const float* xrow = X + (size_t)(row0 + r) * DIM + 2 * g;
        const float* zrow = Z + (size_t)(row0 + r) * DIM + 2 * g;

        v8f acc = {0.f, 0.f, 0.f, 0.f, 0.f, 0.f, 0.f, 0.f};
        v2f a[16];

        // X pass (non-temporal: 256 MB stream, keep L2 for c/xdot)
#pragma unroll
        for (int s = 0; s < 16; ++s)
            a[s] = __builtin_nontemporal_load((const v2f*)(xrow + 4 * s));
#pragma unroll
        for (int s = 0; s < 16; ++s)
            acc = __builtin_amdgcn_wmma_f32_16x16x4_f32(false, a[s], false, wx[s],
                                                        (short)0, acc, false, false);
        // Z pass
#pragma unroll
        for (int s = 0; s < 16; ++s)
            a[s] = __builtin_nontemporal_load((const v2f*)(zrow + 4 * s));
#pragma unroll
        for (int s = 0; s < 16; ++s)
            acc = __builtin_amdgcn_wmma_f32_16x16x4_f32(false, a[s], false, wz[s],
                                                        (short)0, acc, false, false);

        if (r == 0) {           // column 0 = xdot, rows row0+8g .. row0+8g+7
#pragma unroll
            for (int v = 0; v < 8; ++v) xdot_out[row0 + 8 * g + v] = acc[v];
        } else if (r == 1) {    // column 1 = c
#pragma unroll
            for (int v = 0; v < 8; ++v) c_out[row0 + 8 * g + v] = acc[v];
        }
    }
}

// ---------------------------------------------------------------------------
// K2: per-chunk aggregate S_j = sum_{i<CHUNK} G^(CHUNK-1-i) * b[j*CHUNK+i],
//     b[t] = (t==0) ? 0 : c[t-1].  Per-thread Horner over 32 elems, LDS reduce.
// ---------------------------------------------------------------------------
__global__ __launch_bounds__(256) void k2_chunk_agg(
    const float* __restrict__ c, const float* __restrict__ Gp,
    float* __restrict__ S)
{
    __shared__ float red[256];
    const float G = sigmoidf_(Gp[0]);
    const int j = blockIdx.x, tid = threadIdx.x;
    const int i0 = j * CHUNK + tid * LSEG;

    float cprev = (i0 == 0) ? 0.f : c[i0 - 1];
    float p = 0.f;
#pragma unroll
    for (int mm = 0; mm < LSEG / 4; ++mm) {
        const float4 c4 = *(const float4*)(c + i0 + 4 * mm);
        p = p * G + cprev; cprev = c4.x;
        p = p * G + cprev; cprev = c4.y;
        p = p * G + cprev; cprev = c4.z;
        p = p * G + cprev; cprev = c4.w;
    }
    // overall weight: (G^LSEG)^(255-tid)
    red[tid] = p * __powf(G, (float)(CHUNK - (tid + 1) * LSEG));
    __syncthreads();
    for (int s = 128; s > 0; s >>= 1) {
        if (tid < s) red[tid] += red[tid + s];
        __syncthreads();
    }
    if (tid == 0) S[j] = red[0];
}

// ---------------------------------------------------------------------------
// K3: sequential scan over 128 chunk aggregates: P_j = theta_{j*CHUNK-1}.
//     Also emits the [G, eta, zeta] tail of the output tuple.
// ---------------------------------------------------------------------------
__global__ __launch_bounds__(256) void k3_scan_tail(
    const float* __restrict__ Gp, const float* __restrict__ eta,
    const float* __restrict__ zeta, const float* __restrict__ S,
    float* __restrict__ P, float* __restrict__ out_tail)
{
    const int tid = threadIdx.x;
    if (tid == 0) {
        const float G = sigmoidf_(Gp[0]);
        const float A = __powf(G, (float)CHUNK);
        float p = 0.f;
        for (int j = 0; j < NCH; ++j) { P[j] = p; p = A * p + S[j]; }
        out_tail[0] = Gp[0];
    }
    if (tid < DIM) {
        out_tail[1 + tid]        = eta[tid];
        out_tail[1 + DIM + tid]  = zeta[tid];
    }
}

// ---------------------------------------------------------------------------
// K4: apply. Intra-chunk: recompute per-thread Horner partials, scan 256
//     segment entries (thread 0), then Y[t] = theta_t + xdot[t] + c[t].
//     xdot is read from d_out and overwritten in place with Y.
// ---------------------------------------------------------------------------
__global__ __launch_bounds__(256) void k4_apply(
    const float* __restrict__ c, const float* __restrict__ Gp,
    const float* __restrict__ P, float* __restrict__ Y /* in: xdot, out: Y */)
{
    __shared__ float sp[256];
    const float G = sigmoidf_(Gp[0]);
    const int j = blockIdx.x, tid = threadIdx.x;
    const int i0 = j * CHUNK + tid * LSEG;

    const float cprev0 = (i0 == 0) ? 0.f : c[i0 - 1];

    // phase 1: per-segment partial p_tid
    {
        float cprev = cprev0, p = 0.f;
#pragma unroll
        for (int mm = 0; mm < LSEG / 4; ++mm) {
            const float4 c4 = *(const float4*)(c + i0 + 4 * mm);
            p = p * G + cprev; cprev = c4.x;
            p = p * G + cprev; cprev = c4.y;
            p = p * G + cprev; cprev = c4.z;
            p = p * G + cprev; cprev = c4.w;
        }
        sp[tid] = p;
    }
    __syncthreads();

    // phase 2: exclusive scan of segment entry values: E_k = G^L * E_{k-1} + p_{k-1}
    if (tid == 0) {
        const float GL = __powf(G, (float)LSEG);
        float e = P[j];
        for (int k = 0; k < 256; ++k) {
            const float pk = sp[k];
            sp[k] = e;             // entry theta for segment k
            e = GL * e + pk;
        }
    }
    __syncthreads();

    // phase 3: emit Y
    float th = sp[tid];
    float cprev = cprev0;
#pragma unroll
    for (int mm = 0; mm < LSEG / 4; ++mm) {
        const float4 c4 = *(const float4*)(c + i0 + 4 * mm);
        float4 y4 = *(const float4*)(Y + i0 + 4 * mm);   // xdot
        th = G * th + cprev; y4.x += th + c4.x; cprev = c4.x;
        th = G * th + cprev; y4.y += th + c4.y; cprev = c4.y;
        th = G * th + cprev; y4.z += th + c4.z; cprev = c4.z;
        th = G * th + cprev; y4.w += th + c4.w; cprev = c4.w;
        *(float4*)(Y + i0 + 4 * mm) = y4;
    }
}

extern "C" void kernel_launch(void* const* d_in, const int* in_sizes, int n_in,
                              void* d_out, int out_size, void* d_ws, size_t ws_size,
                              hipStream_t stream)
{
    (void)in_sizes; (void)n_in; (void)out_size; (void)ws_size;
    const float* X    = (const float*)d_in[0];
    const float* Z    = (const float*)d_in[1];
    const float* Gp   = (const float*)d_in[2];
    const float* eta  = (const float*)d_in[3];
    const float* zeta = (const float*)d_in[4];
    float* out = (float*)d_out;
    float* ws  = (float*)d_ws;

    float* c = ws;               // T_TOTAL floats
    float* S = ws + T_TOTAL;     // NCH floats
    float* P = S + NCH;          // NCH floats

    // 512 blocks * 8 waves * 16 tiles * 16 rows = 1,048,576 rows
    k1_matvec_wmma<<<512, 256, 0, stream>>>(X, Z, eta, zeta, out, c);
    k2_chunk_agg  <<<NCH, 256, 0, stream>>>(c, Gp, S);
    k3_scan_tail  <<<1,   256, 0, stream>>>(Gp, eta, zeta, S, P, out + T_TOTAL);
    k4_apply      <<<NCH, 256, 0, stream>>>(c, Gp, P, out);
}